// HouseholderMultiplier_78752520339509
// MI455X (gfx1250) — compile-verified
//
#include <hip/hip_runtime.h>
#include <hip/hip_bf16.h>

// Block-diagonal (grouped Hadamard) transform:
//   y[n, g*64 + h] = sum_k x[n, g*64 + k] * H[g, k, h]
// HBM-bound (AI ~16 flop/byte, floor ~23us at 23.3 TB/s). Math runs on
// V_WMMA_F32_16X16X4_F32 (exact fp32). H (1 MB) is pre-packed into the WMMA
// B-fragment register image so B loads are coalesced b128 instead of 128
// scalar b32 per wave.

typedef __attribute__((ext_vector_type(2))) float v2f;
typedef __attribute__((ext_vector_type(4))) float v4f;
typedef __attribute__((ext_vector_type(8))) float v8f;

#define DIM    4096
#define GROUP  64
#define NG     (DIM / GROUP)  // 64 hadamard groups
#define GPB    8              // groups per HIP block (one per wave)
#define ROWF   (GPB * GROUP)  // 512 floats of x per token row per block
#define ROWPAD (ROWF + 4)     // 516 dwords -> bank-conflict-free A reads

// ---------------------------------------------------------------------------
// Pre-pack H[g] (64x64 row-major) into per-lane WMMA B fragments:
//   Hp[(((g*4 + j)*8 + t)*32 + l)*4 + c] = H[g][k][n]
//   k = 8t + 4*(c>>1) + 2*(l>>4) + (c&1),  n = j*16 + (l&15)
// One v4f per lane covers k-steps s=2t and s=2t+1.
// ---------------------------------------------------------------------------
__global__ __launch_bounds__(256) void pack_H_kernel(const float* __restrict__ H,
                                                     float* __restrict__ Hp) {
  const int i = blockIdx.x * 256 + threadIdx.x;  // [0, 64*4096)
  const int c = i & 3;
  const int l = (i >> 2) & 31;
  const int t = (i >> 7) & 7;
  const int j = (i >> 10) & 3;
  const int g = i >> 12;
  const int k = 8 * t + 4 * (c >> 1) + 2 * (l >> 4) + (c & 1);
  const int n = j * 16 + (l & 15);
  Hp[i] = H[((size_t)g * GROUP + k) * GROUP + n];
}

// ---------------------------------------------------------------------------
// Main kernel: packed-B path.
// ---------------------------------------------------------------------------
__global__ __launch_bounds__(256) void hadamard_wmma_kernel(
    const float* __restrict__ X, const float* __restrict__ Hp,
    float* __restrict__ Y) {
  __shared__ float lds[16 * ROWPAD];  // 33 KB

  const int t     = threadIdx.x;
  const int tile  = blockIdx.x;       // 16-token tile index
  const int gbase = blockIdx.y * GPB;

  // ---- Stage 16 tokens x 512 columns of x into LDS (coalesced, NT) ----
  {
    const int col4  = (t & 127) * 4;  // 128 threads cover one 512-float row
    const int rbase = t >> 7;         // 0 or 1
    const float* src = X + (size_t)(tile * 16) * DIM + (size_t)gbase * GROUP;
#pragma unroll
    for (int i = 0; i < 8; ++i) {
      const int row = i * 2 + rbase;
      const v4f v =
          __builtin_nontemporal_load((const v4f*)(src + (size_t)row * DIM + col4));
      *(v4f*)(&lds[row * ROWPAD + col4]) = v;
    }
  }
  __syncthreads();

  const int lane = t & 31;
  const int w    = t >> 5;       // wave id -> which group
  const int g    = gbase + w;
  const int lo   = lane & 15;    // M (A) / N (B, D)
  const int hi   = lane >> 4;    // lane-half selects K pair (A,B) / M half (D)

  // ---- A fragments: 16 k-steps of the 16x4 f32 A layout, kept in VGPRs ----
  v2f a[16];
#pragma unroll
  for (int s = 0; s < 16; ++s) {
    const int col = w * GROUP + 4 * s + 2 * hi;
    a[s] = *(const v2f*)(&lds[lo * ROWPAD + col]);
  }

  float* Yg = Y + (size_t)(tile * 16) * DIM + (size_t)g * GROUP;

#pragma unroll
  for (int j = 0; j < 4; ++j) {  // four 16-wide N tiles cover h = 0..63
    v8f acc = {};
    const float* Bp = Hp + ((size_t)(g * 4 + j) * 8) * 128 + lane * 4;
#pragma unroll
    for (int tt = 0; tt < 8; ++tt) {  // two k-steps per b128 load
      const v4f bb = *(const v4f*)(Bp + tt * 128);
      const v2f b0 = {bb.x, bb.y};
      const v2f b1 = {bb.z, bb.w};
      acc = __builtin_amdgcn_wmma_f32_16x16x4_f32(
          false, a[2 * tt], false, b0, (short)0, acc, false, false);
      acc = __builtin_amdgcn_wmma_f32_16x16x4_f32(
          false, a[2 * tt + 1], false, b1, (short)0, acc, false, false);
    }
    // D 16x16 f32: lane%16 = N; VGPR v -> M = v + 8*(lane/16)
#pragma unroll
    for (int v = 0; v < 8; ++v) {
      const int m = 8 * hi + v;
      __builtin_nontemporal_store(acc[v], &Yg[(size_t)m * DIM + j * 16 + lo]);
    }
  }
}

// ---------------------------------------------------------------------------
// Fallback (no workspace): B loaded directly from row-major H (scalar loads).
// ---------------------------------------------------------------------------
__global__ __launch_bounds__(256) void hadamard_wmma_fallback(
    const float* __restrict__ X, const float* __restrict__ H,
    float* __restrict__ Y) {
  __shared__ float lds[16 * ROWPAD];

  const int t     = threadIdx.x;
  const int tile  = blockIdx.x;
  const int gbase = blockIdx.y * GPB;
  {
    const int col4  = (t & 127) * 4;
    const int rbase = t >> 7;
    const float* src = X + (size_t)(tile * 16) * DIM + (size_t)gbase * GROUP;
#pragma unroll
    for (int i = 0; i < 8; ++i) {
      const int row = i * 2 + rbase;
      const v4f v =
          __builtin_nontemporal_load((const v4f*)(src + (size_t)row * DIM + col4));
      *(v4f*)(&lds[row * ROWPAD + col4]) = v;
    }
  }
  __syncthreads();

  const int lane = t & 31;
  const int w    = t >> 5;
  const int g    = gbase + w;
  const int lo   = lane & 15;
  const int hi   = lane >> 4;

  v2f a[16];
#pragma unroll
  for (int s = 0; s < 16; ++s) {
    const int col = w * GROUP + 4 * s + 2 * hi;
    a[s] = *(const v2f*)(&lds[lo * ROWPAD + col]);
  }

  const float* Hg = H + (size_t)g * GROUP * GROUP;
  float* Yg = Y + (size_t)(tile * 16) * DIM + (size_t)g * GROUP;

#pragma unroll
  for (int j = 0; j < 4; ++j) {
    v8f acc = {};
#pragma unroll
    for (int s = 0; s < 16; ++s) {
      const int k = 4 * s + 2 * hi;
      v2f b;
      b.x = Hg[(size_t)k * GROUP + j * 16 + lo];
      b.y = Hg[(size_t)(k + 1) * GROUP + j * 16 + lo];
      acc = __builtin_amdgcn_wmma_f32_16x16x4_f32(
          false, a[s], false, b, (short)0, acc, false, false);
    }
#pragma unroll
    for (int v = 0; v < 8; ++v) {
      const int m = 8 * hi + v;
      __builtin_nontemporal_store(acc[v], &Yg[(size_t)m * DIM + j * 16 + lo]);
    }
  }
}

extern "C" void kernel_launch(void* const* d_in, const int* in_sizes, int n_in,
                              void* d_out, int out_size, void* d_ws, size_t ws_size,
                              hipStream_t stream) {
  const float* x = (const float*)d_in[0];  // [4,4096,4096] f32
  const float* H = (const float*)d_in[1];  // [64,64,64]   f32
  float* out = (float*)d_out;

  const int n_tokens = in_sizes[0] / DIM;          // 16384
  dim3 grid(n_tokens / 16, (DIM / GROUP) / GPB);   // (1024, 8)

  const size_t packed_bytes = (size_t)NG * GROUP * GROUP * sizeof(float);  // 1 MB
  if (ws_size >= packed_bytes) {
    float* Hp = (float*)d_ws;
    pack_H_kernel<<<(NG * GROUP * GROUP) / 256, 256, 0, stream>>>(H, Hp);
    hadamard_wmma_kernel<<<grid, 256, 0, stream>>>(x, Hp, out);
  } else {
    hadamard_wmma_fallback<<<grid, 256, 0, stream>>>(x, H, out);
  }
}